// WignerOrgan_73778948211236
// MI455X (gfx1250) — compile-verified
//
#include <hip/hip_runtime.h>
#include <math.h>

typedef __attribute__((ext_vector_type(2))) float v2f;
typedef __attribute__((ext_vector_type(8))) float v8f;

#define NN 4096
#define BLK_I 256     // block tile rows (i)
#define BLK_M 128     // block tile cols (m)
#define BK 16
#define A_STRIDE 17   // 256 rows x 16 k, padded: gcd(17,64)=1 -> conflict-free column reads
#define B_STRIDE 128  // 16 k-rows x 128 m; consecutive-m reads are conflict-free

__device__ __forceinline__ unsigned float_orderable(float f) {
    unsigned u = __float_as_uint(f);
    return (u & 0x80000000u) ? ~u : (u | 0x80000000u);
}

// ---------------------------------------------------------------------------
// Init: mean(x), dx, dp, and seeded reduction accumulators in ws.
// ws layout (32-bit slots): [0]=xmean [1]=dx [2]=dp [3]=min_bits [4]=negsum [5]=abssum
// ---------------------------------------------------------------------------
__global__ void wig_init(const float* __restrict__ x, const float* __restrict__ p,
                         float* __restrict__ wsf, unsigned* __restrict__ wsu) {
    __shared__ float red[256];
    const int tid = threadIdx.x;
    float s = 0.f;
    for (int q = tid; q < NN; q += 256) s += x[q];
    red[tid] = s;
    __syncthreads();
    for (int k = 128; k > 0; k >>= 1) {
        if (tid < k) red[tid] += red[tid + k];
        __syncthreads();
    }
    if (tid == 0) {
        wsf[0] = red[0] * (1.0f / (float)NN);
        wsf[1] = x[1] - x[0];
        wsf[2] = p[1] - p[0];
        wsu[3] = 0xFFFFFFFFu;   // orderable(+inf) sentinel for min
        wsf[4] = 0.f;
        wsf[5] = 0.f;
    }
}

// ---------------------------------------------------------------------------
// Main: W[i,m] = sum_j ( Cr[i,j]*cos(2 p_m y_j) + (-Ci[i,j])*sin(2 p_m y_j) ) * dx/pi
// Block = 256x128 tile of W, 8 waves (4 i x 2 m), wave = 64x64 = 4x4 WMMA tiles.
// Per k-substep: 32 f32 WMMAs from 16 fragment loads (2:1 reuse).
// ---------------------------------------------------------------------------
__global__ __launch_bounds__(256)
void wig_main(const float* __restrict__ psiR, const float* __restrict__ psiI,
              const float* __restrict__ xg, const float* __restrict__ pg,
              float* __restrict__ out, float* __restrict__ wsf,
              unsigned* __restrict__ wsu) {
    __shared__ float sAcr[BLK_I * A_STRIDE];   // C.real tile     [i][k]
    __shared__ float sAnc[BLK_I * A_STRIDE];   // -C.imag tile    [i][k]
    __shared__ float sBr[BK * B_STRIDE];       // cos phases      [k][m]
    __shared__ float sBi[BK * B_STRIDE];       // sin phases      [k][m]
    __shared__ float sNeg, sAbs;
    __shared__ unsigned sMin;

    const int tid  = threadIdx.x;
    const int lane = tid & 31;
    const int wid  = tid >> 5;
    const int wi   = (wid >> 1) * 64;          // wave row offset inside block (0..192)
    const int wm   = (wid & 1) * 64;           // wave col offset inside block (0,64)
    const int iBase = blockIdx.y * BLK_I;
    const int mBase = blockIdx.x * BLK_M;
    const int lhalf = lane >> 4;               // 0: lanes 0-15, 1: lanes 16-31
    const int l15   = lane & 15;

    if (tid == 0) { sNeg = 0.f; sAbs = 0.f; sMin = 0xFFFFFFFFu; }
    const float xmean = wsf[0];
    const float dxv   = wsf[1];
    __syncthreads();

    v8f acc[4][4];
    const v8f vzero = {};
#pragma unroll
    for (int t = 0; t < 4; ++t)
#pragma unroll
        for (int u = 0; u < 4; ++u) acc[t][u] = vzero;

    for (int j0 = 0; j0 < NN; j0 += BK) {
        // ---- stage A tile: Cr and -Ci for 256 rows x 16 k (psi gathers via cache)
        for (int q = tid; q < BLK_I * BK; q += 256) {
            const int il = q >> 4, jl = q & 15;
            const int ig = iBase + il, jg = j0 + jl;
            const unsigned ia = (unsigned)(jg - ig) & (NN - 1);
            const unsigned ib = (unsigned)(jg + ig) & (NN - 1);
            const float ar = __ldg(psiR + ia), ai = __ldg(psiI + ia);
            const float br = __ldg(psiR + ib), bi = __ldg(psiI + ib);
            sAcr[il * A_STRIDE + jl] = ar * br + ai * bi;   //  Re(psi_a * conj(psi_b))
            sAnc[il * A_STRIDE + jl] = ar * bi - ai * br;   // -Im(psi_a * conj(psi_b))
        }
        // ---- stage B tile: cos/sin(2 p_m y_j) for 16 k x 128 m
        for (int q = tid; q < BK * BLK_M; q += 256) {
            const int kl = q >> 7, ml = q & 127;
            const float arg = 2.0f * __ldg(pg + mBase + ml) * (__ldg(xg + j0 + kl) - xmean);
            float r = arg * 0.15915494309189535f;  // radians -> revolutions
            r -= rintf(r);                         // range-reduce to [-0.5, 0.5]
            sBr[kl * B_STRIDE + ml] = __builtin_amdgcn_cosf(r);
            sBi[kl * B_STRIDE + ml] = __builtin_amdgcn_sinf(r);
        }
        __syncthreads();

#pragma unroll
        for (int kk = 0; kk < 4; ++kk) {
            const int kb = kk * 4 + (lhalf << 1);  // per-lane K base per ISA A/B layout
            v2f aC[4], aS[4], bC[4], bS[4];
#pragma unroll
            for (int t = 0; t < 4; ++t) {
                const int base = (wi + t * 16 + l15) * A_STRIDE + kb;
                v2f ac; ac.x = sAcr[base]; ac.y = sAcr[base + 1]; aC[t] = ac;
                v2f as; as.x = sAnc[base]; as.y = sAnc[base + 1]; aS[t] = as;
            }
#pragma unroll
            for (int u = 0; u < 4; ++u) {
                const int col = wm + u * 16 + l15;
                v2f bc; bc.x = sBr[kb * B_STRIDE + col]; bc.y = sBr[(kb + 1) * B_STRIDE + col];
                v2f bs; bs.x = sBi[kb * B_STRIDE + col]; bs.y = sBi[(kb + 1) * B_STRIDE + col];
                bC[u] = bc; bS[u] = bs;
            }
            // 16 independent cos-WMMAs, then 16 sin-WMMAs (keeps D->C chains spaced)
#pragma unroll
            for (int t = 0; t < 4; ++t)
#pragma unroll
                for (int u = 0; u < 4; ++u)
                    acc[t][u] = __builtin_amdgcn_wmma_f32_16x16x4_f32(
                        false, aC[t], false, bC[u], (short)0, acc[t][u], false, false);
#pragma unroll
            for (int t = 0; t < 4; ++t)
#pragma unroll
                for (int u = 0; u < 4; ++u)
                    acc[t][u] = __builtin_amdgcn_wmma_f32_16x16x4_f32(
                        false, aS[t], false, bS[u], (short)0, acc[t][u], false, false);
        }
        __syncthreads();
    }

    // ---- epilogue: scale, store W, local reductions
    const float scale = dxv * 0.31830988618379067f;  // dx / (pi * hbar)
    float lmin = 3.4e38f, lneg = 0.f, labs = 0.f;
#pragma unroll
    for (int t = 0; t < 4; ++t)
#pragma unroll
        for (int u = 0; u < 4; ++u)
#pragma unroll
            for (int r = 0; r < 8; ++r) {
                const int ig = iBase + wi + t * 16 + r + (lhalf << 3);
                const int mg = mBase + wm + u * 16 + l15;
                const float w = acc[t][u][r] * scale;
                out[(size_t)ig * NN + mg] = w;
                lmin = fminf(lmin, w);
                labs += fabsf(w);
                lneg += fmaxf(-w, 0.f);
            }

    atomicMin(&sMin, float_orderable(lmin));
    atomicAdd(&sNeg, lneg);
    atomicAdd(&sAbs, labs);
    __syncthreads();
    if (tid == 0) {
        atomicMin(&wsu[3], sMin);
        atomicAdd(&wsf[4], sNeg);
        atomicAdd(&wsf[5], sAbs);
    }
}

// ---------------------------------------------------------------------------
// Finalize: 3 trailing scalars (W.min, neg_mass, neg_ratio)
// ---------------------------------------------------------------------------
__global__ void wig_fin(const float* __restrict__ wsf, const unsigned* __restrict__ wsu,
                        float* __restrict__ out3) {
    const unsigned u = wsu[3];
    const float wmin = __uint_as_float((u & 0x80000000u) ? (u ^ 0x80000000u) : ~u);
    const float dx = wsf[1], dp = wsf[2];
    const float neg = wsf[4] * dx * dp;
    const float tot = wsf[5] * dx * dp + 1e-12f;
    out3[0] = wmin;
    out3[1] = neg;
    out3[2] = neg / tot;
}

extern "C" void kernel_launch(void* const* d_in, const int* in_sizes, int n_in,
                              void* d_out, int out_size, void* d_ws, size_t ws_size,
                              hipStream_t stream) {
    const float* psiR = (const float*)d_in[0];
    const float* psiI = (const float*)d_in[1];
    const float* x    = (const float*)d_in[2];
    const float* p    = (const float*)d_in[3];
    float*    out = (float*)d_out;
    float*    wsf = (float*)d_ws;
    unsigned* wsu = (unsigned*)d_ws;

    wig_init<<<dim3(1), dim3(256), 0, stream>>>(x, p, wsf, wsu);
    dim3 grid(NN / BLK_M, NN / BLK_I);
    wig_main<<<grid, dim3(256), 0, stream>>>(psiR, psiI, x, p, out, wsf, wsu);
    wig_fin<<<dim3(1), dim3(1), 0, stream>>>(wsf, wsu, out + (size_t)NN * NN);
}